// ContrastiveMoCoKnnBert_29764123361920
// MI455X (gfx1250) — compile-verified
//
#include <hip/hip_runtime.h>
#include <hip/hip_bf16.h>

typedef __attribute__((ext_vector_type(2))) float v2f;
typedef __attribute__((ext_vector_type(8))) float v8f;

#define HDIM 768
#define BDIM 32
#define KQ   65536

// ---------- float <-> descending-order radix key ----------
__device__ __forceinline__ unsigned descKey(float f) {
    unsigned u = __float_as_uint(f);
    unsigned asc = (u & 0x80000000u) ? ~u : (u | 0x80000000u); // ascending-ordered
    return ~asc;                                               // ascending key == descending float
}
__device__ __forceinline__ float descDecode(unsigned key) {
    unsigned asc = ~key;
    unsigned u = (asc & 0x80000000u) ? (asc ^ 0x80000000u) : ~asc;
    return __uint_as_float(u);
}
#define NEGINF_KEY 0xFF800000u  // == descKey(-inf)

// ---------- fp32 WMMA NT GEMM: C[m][n] = sum_k A[m][k]*W[n][k], M=32, K=768 ----------
// block = 256 threads (8 waves). Tile: M=32, N=64 per block, K chunked by 32.
// Staging uses GLOBAL_LOAD_ASYNC_TO_LDS_B128 (ASYNCcnt) - no VGPR round-trip.
// act: 0 = +bias, 1 = tanh(+bias), 2 = raw
__global__ __launch_bounds__(256) void gemm_nt_32xK(const float* __restrict__ A,
                                                    const float* __restrict__ W,
                                                    const float* __restrict__ bias,
                                                    float* __restrict__ C,
                                                    int Ntot, int act) {
    // row stride 36 floats: 16B-aligned rows for async B128, 16 distinct banks
    __shared__ float smem[32 * 36 + 64 * 36];
    float (*sA)[36] = (float(*)[36])smem;
    float (*sB)[36] = (float(*)[36])(smem + 32 * 36);

    const int tid   = threadIdx.x;
    const int nbase = blockIdx.x * 64;
    const int wave = tid >> 5, lane = tid & 31;
    const int half = lane >> 4, l = lane & 15;
    const int mBase = (wave & 1) * 16;
    const int nSub  = (wave >> 1) * 16;
    const int lr = tid >> 3;        // 0..31 loader row
    const int lc = (tid & 7) * 4;   // 0..28 loader col (x4 floats)

    const unsigned long long aBase = (unsigned long long)A;
    const unsigned long long wBase = (unsigned long long)W;
    const unsigned ldsBase = (unsigned)(unsigned long long)smem; // addr[31:0] = LDS offset
    const unsigned ldsA  = ldsBase + (unsigned)(lr * 36 + lc) * 4u;
    const unsigned ldsB0 = ldsBase + (unsigned)((32 + lr) * 36 + lc) * 4u;
    const unsigned ldsB1 = ldsBase + (unsigned)((32 + lr + 32) * 36 + lc) * 4u;

    v8f acc = {};
    for (int k0 = 0; k0 < HDIM; k0 += 32) {
        // async copy: A chunk 32x32, W chunk 64x32, straight into LDS
        unsigned gA  = (unsigned)(lr * HDIM + k0 + lc) * 4u;
        unsigned gB0 = (unsigned)((nbase + lr) * HDIM + k0 + lc) * 4u;
        unsigned gB1 = (unsigned)((nbase + lr + 32) * HDIM + k0 + lc) * 4u;
        asm volatile("global_load_async_to_lds_b128 %0, %1, %2"
                     :: "v"(ldsA), "v"(gA), "s"(aBase) : "memory");
        asm volatile("global_load_async_to_lds_b128 %0, %1, %2"
                     :: "v"(ldsB0), "v"(gB0), "s"(wBase) : "memory");
        asm volatile("global_load_async_to_lds_b128 %0, %1, %2"
                     :: "v"(ldsB1), "v"(gB1), "s"(wBase) : "memory");
        if (k0 + 32 < HDIM) { // pull next W chunk toward L2 (global_prefetch_b8)
            __builtin_prefetch(W + (size_t)(nbase + lr) * HDIM + k0 + 32 + lc, 0, 1);
            __builtin_prefetch(W + (size_t)(nbase + lr + 32) * HDIM + k0 + 32 + lc, 0, 1);
        }
        asm volatile("s_wait_asynccnt 0" ::: "memory"); // this wave's async copies done
        __syncthreads();                                // all waves' copies visible
#pragma unroll
        for (int kk = 0; kk < 32; kk += 4) {
            v2f a, b;
            // A 16x4: lanes 0-15 hold K=kk..kk+1, lanes 16-31 hold K=kk+2..kk+3
            a.x = sA[mBase + l][kk + 2 * half];
            a.y = sA[mBase + l][kk + 2 * half + 1];
            // B 4x16 mirrored over N
            b.x = sB[nSub + l][kk + 2 * half];
            b.y = sB[nSub + l][kk + 2 * half + 1];
            acc = __builtin_amdgcn_wmma_f32_16x16x4_f32(false, a, false, b,
                                                        (short)0, acc, false, false);
        }
        __syncthreads();
    }
    // D layout: VGPR i -> lanes0-15: M=i, lanes16-31: M=8+i; N = lane&15
    const int col = nbase + nSub + l;
#pragma unroll
    for (int i = 0; i < 8; ++i) {
        int row = mBase + half * 8 + i;
        float v = acc[i];
        if (act != 2) v += bias[col];
        if (act == 1) v = tanhf(v);
        C[(size_t)row * Ntot + col] = v;
    }
}

// ---------- per-row L2 normalize (32 rows x 768) ----------
__global__ __launch_bounds__(256) void row_normalize(const float* __restrict__ x,
                                                     float* __restrict__ out) {
    __shared__ float red[256];
    const int b = blockIdx.x, tid = threadIdx.x;
    float s = 0.f;
    for (int h = tid; h < HDIM; h += 256) { float v = x[b * HDIM + h]; s += v * v; }
    red[tid] = s; __syncthreads();
    for (int st = 128; st > 0; st >>= 1) {
        if (tid < st) red[tid] += red[tid + st];
        __syncthreads();
    }
    const float inv = 1.0f / sqrtf(red[0]);
    for (int h = tid; h < HDIM; h += 256) out[b * HDIM + h] = x[b * HDIM + h] * inv;
}

// ---------- build pos/neg radix keys from cos_sim + label mask ----------
__global__ __launch_bounds__(256) void build_keys(const float* __restrict__ cosb,
                                                  const int* __restrict__ labels,
                                                  const int* __restrict__ lqueue,
                                                  unsigned* __restrict__ posK,
                                                  unsigned* __restrict__ negK) {
    size_t idx = (size_t)blockIdx.x * 256 + threadIdx.x; // < 32*65536
    int b = (int)(idx >> 16);
    int k = (int)(idx & 65535);
    unsigned key = descKey(cosb[idx]);
    bool match = (labels[b] == lqueue[k]);
    posK[idx] = match ? key : NEGINF_KEY;
    negK[idx] = match ? NEGINF_KEY : key;
}

// ---------- 4x8-bit LDS-histogram radix sort of one 64K slice per block ----------
__global__ __launch_bounds__(256) void radix_sort64k(unsigned* __restrict__ bufA,
                                                     unsigned* __restrict__ bufB) {
    __shared__ unsigned hist[256];
    __shared__ unsigned scanb[256];
    const size_t off = (size_t)blockIdx.x << 16;
    unsigned* a = bufA + off;
    unsigned* bp = bufB + off;
    for (int p = 0; p < 4; ++p) {
        unsigned* in  = (p & 1) ? bp : a;
        unsigned* out = (p & 1) ? a : bp;
        const int sh = p * 8;
        hist[threadIdx.x] = 0u;
        __syncthreads();
        for (int i = threadIdx.x; i < KQ; i += 256)
            atomicAdd(&hist[(in[i] >> sh) & 255u], 1u);
        __syncthreads();
        if (threadIdx.x == 0) {
            unsigned s = 0;
            for (int j = 0; j < 256; ++j) { unsigned c = hist[j]; scanb[j] = s; s += c; }
        }
        __syncthreads();
        hist[threadIdx.x] = scanb[threadIdx.x];
        __syncthreads();
        for (int i = threadIdx.x; i < KQ; i += 256) {
            unsigned v = in[i];
            unsigned pos = atomicAdd(&hist[(v >> sh) & 255u], 1u);
            out[pos] = v;
        }
        __syncthreads();
    }
    // 4 passes: A->B->A->B->A : result in bufA
}

// ---------- pos_min = min_b count(label_queue == labels[b]) ----------
__global__ __launch_bounds__(256) void compute_posmin(const int* __restrict__ labels,
                                                      const int* __restrict__ lqueue,
                                                      int* __restrict__ posmin) {
    __shared__ int c0s[256];
    const int tid = threadIdx.x;
    int c = 0;
    for (int k = tid; k < KQ; k += 256) c += (lqueue[k] == 0) ? 1 : 0;
    c0s[tid] = c; __syncthreads();
    for (int st = 128; st > 0; st >>= 1) {
        if (tid < st) c0s[tid] += c0s[tid + st];
        __syncthreads();
    }
    if (tid == 0) {
        int count0 = c0s[0];
        int pm = 1 << 30;
        for (int b = 0; b < BDIM; ++b) {
            int pc = (labels[b] == 0) ? count0 : (KQ - count0);
            pm = pc < pm ? pc : pm;
        }
        *posmin = pm;
    }
}

// ---------- emit logits: row (b*50+r): [pos_r, neg_sorted[0..neg_min)]/T ----------
__global__ __launch_bounds__(256) void emit_logits(const unsigned* __restrict__ posK,
                                                   const unsigned* __restrict__ negK,
                                                   const int* __restrict__ posmin_p,
                                                   int neg_min, float* __restrict__ out) {
    const int b = blockIdx.x / 50;
    const int r = blockIdx.x % 50;
    const int pm = *posmin_p;
    const size_t rowBase = (size_t)blockIdx.x * (size_t)(neg_min + 1);
    const float invT = 1.0f / 0.07f;
    if (threadIdx.x == 0) {
        int pi = (r < 25) ? r : (pm - 25 + (r - 25));
        out[rowBase] = descDecode(posK[(size_t)b * KQ + pi]) * invT;
    }
    for (int j = threadIdx.x; j < neg_min; j += 256)
        out[rowBase + 1 + j] = descDecode(negK[(size_t)b * KQ + j]) * invT;
}

extern "C" void kernel_launch(void* const* d_in, const int* in_sizes, int n_in,
                              void* d_out, int out_size, void* d_ws, size_t ws_size,
                              hipStream_t stream) {
    const float* q       = (const float*)d_in[0];
    const float* dense_w = (const float*)d_in[1];
    const float* dense_b = (const float*)d_in[2];
    const float* out_w   = (const float*)d_in[3];
    const float* out_b   = (const float*)d_in[4];
    const int*   labels  = (const int*)d_in[5];
    const int*   lqueue  = (const int*)d_in[6];
    const float* fq      = (const float*)d_in[7];
    float* out = (float*)d_out;

    char* w = (char*)d_ws;
    float* hidden = (float*)w;  w += (size_t)BDIM * HDIM * 4;
    float* xbuf   = (float*)w;  w += (size_t)BDIM * HDIM * 4;
    float* lqv    = (float*)w;  w += (size_t)BDIM * HDIM * 4;
    float* cosb   = (float*)w;  w += (size_t)BDIM * KQ * 4;
    unsigned* keysA = (unsigned*)w; w += (size_t)2 * BDIM * KQ * 4;
    unsigned* keysB = (unsigned*)w; w += (size_t)2 * BDIM * KQ * 4;
    int* posmin = (int*)w;

    const int neg_min = out_size / (BDIM * 50) - 1;

    // MLP head: hidden = tanh(q @ dense_w^T + dense_b); x = hidden @ out_w^T + out_b
    gemm_nt_32xK<<<HDIM / 64, 256, 0, stream>>>(q, dense_w, dense_b, hidden, HDIM, 1);
    gemm_nt_32xK<<<HDIM / 64, 256, 0, stream>>>(hidden, out_w, out_b, xbuf, HDIM, 0);
    row_normalize<<<BDIM, 256, 0, stream>>>(xbuf, lqv);

    // cos_sim = liner_q @ feature_queue^T  (HBM-bound: 201 MB streamed once)
    gemm_nt_32xK<<<KQ / 64, 256, 0, stream>>>(lqv, fq, nullptr, cosb, KQ, 2);

    // selection pipeline
    build_keys<<<(BDIM * KQ) / 256, 256, 0, stream>>>(cosb, labels, lqueue,
                                                      keysA, keysA + (size_t)BDIM * KQ);
    compute_posmin<<<1, 256, 0, stream>>>(labels, lqueue, posmin);
    radix_sort64k<<<2 * BDIM, 256, 0, stream>>>(keysA, keysB);
    emit_logits<<<BDIM * 50, 256, 0, stream>>>(keysA, keysA + (size_t)BDIM * KQ,
                                               posmin, neg_min, out);
}